// RotationallySymmetricQuantizedHeightMapDOELayer_19490561590229
// MI455X (gfx1250) — compile-verified
//
#include <hip/hip_runtime.h>
#include <stdint.h>

#define RADIUS 512
#define DIAM 1024
#define N_WL 31
#define NPIXEL (DIAM * DIAM)
#define N_ELEM (NPIXEL * N_WL)              /* 32,505,856 */
#define ELEMS_PER_BLOCK 8192
#define THREADS 256
#define F4_PER_BLOCK (ELEMS_PER_BLOCK / 4)  /* 2048 */
#define ITERS (F4_PER_BLOCK / THREADS)      /* 8 */
#define NBLOCKS (N_ELEM / ELEMS_PER_BLOCK)  /* 3968, exact */
#define MAX_PIX 272                         /* >= ceil(8192/31)+2 */

typedef float v4f __attribute__((ext_vector_type(4)));
typedef float v2f __attribute__((ext_vector_type(2)));

/* ---- CDNA5 async global->LDS path (guarded; falls back to plain loads) ---- */
#if defined(__AMDGCN__) && defined(__has_builtin)
#if __has_builtin(__builtin_amdgcn_global_load_async_to_lds_b32) && \
    __has_builtin(__builtin_amdgcn_s_wait_asynccnt)
#define USE_ASYNC_LDS 1
#endif
#endif

typedef __attribute__((address_space(1))) int glb_int;
typedef __attribute__((address_space(3))) int lds_int;

__device__ __forceinline__ glb_int *to_glb(const void *p) {
  return (glb_int *)(uintptr_t)p;
}
__device__ __forceinline__ lds_int *to_lds(const void *p) {
  /* LDS flat addresses keep the byte offset in the low 32 bits */
  return (lds_int *)(uint32_t)(uintptr_t)p;
}

/* ---------------- Kernel 1: tiny prep (table[512], kdn[31]) --------------- */
__global__ void doe_prep_kernel(const float *__restrict__ hmw,
                                const float *__restrict__ wavelength,
                                float *__restrict__ ws) {
  const int t = threadIdx.x;
  if (t < RADIUS) {
    float w = hmw[t];
    w = fminf(fmaxf(w, -1.0f), 1.0f);
    const float normed = (w + 1.0f) * 0.5f;
    const float lam0 = 7e-07f;
    const float n0m1 = 0.5f + 4e-15f / (lam0 * lam0); /* n(lam0)-1 */
    const float q_base_height = lam0 / n0m1;
    ws[t] = 0.002f - q_base_height * normed;
  }
  if (t < N_WL) {
    const float lam = wavelength[t];
    const float dn = 0.5f + 4e-15f / (lam * lam); /* n(lam)-1 */
    const float k = 6.28318530717958647692f / lam;
    ws[RADIUS + t] = k * dn;
  }
}

/* ---------------- Kernel 2: streaming phase-rotation kernel --------------- */
__global__ void __launch_bounds__(THREADS)
doe_main_kernel(const float *__restrict__ fieldR,
                const float *__restrict__ fieldI,
                const float *__restrict__ noise,
                const float *__restrict__ rdist,
                const float *__restrict__ aperture,
                const float *__restrict__ ws,
                float *__restrict__ outR,
                float *__restrict__ outI) {
  __shared__ v2f sPix[MAX_PIX];  /* {hmap+noise, aperture} per pixel */
  __shared__ float sK[32];       /* k * delta_n per wavelength */
#ifdef USE_ASYNC_LDS
  __shared__ float sR[MAX_PIX];
  __shared__ float sN[MAX_PIX];
  __shared__ float sA[MAX_PIX];
#endif

  const unsigned tid = threadIdx.x;
  const unsigned eStart = (unsigned)blockIdx.x * ELEMS_PER_BLOCK;
  const unsigned pixStart = eStart / 31u;
  const unsigned pixEnd = (eStart + ELEMS_PER_BLOCK - 1u) / 31u;
  const unsigned nPix = pixEnd - pixStart + 1u;

#ifdef USE_ASYNC_LDS
  /* Stage raw per-pixel data via CDNA5 async DMA (ASYNCcnt-tracked). */
  for (unsigned p = tid; p < nPix; p += THREADS) {
    const unsigned gp = pixStart + p;
    __builtin_amdgcn_global_load_async_to_lds_b32(to_glb(rdist + gp),
                                                  to_lds(&sR[p]), 0, 0);
    __builtin_amdgcn_global_load_async_to_lds_b32(to_glb(noise + gp),
                                                  to_lds(&sN[p]), 0, 0);
    __builtin_amdgcn_global_load_async_to_lds_b32(to_glb(aperture + gp),
                                                  to_lds(&sA[p]), 0, 0);
  }
  if (tid < N_WL) {
    __builtin_amdgcn_global_load_async_to_lds_b32(to_glb(ws + RADIUS + tid),
                                                  to_lds(&sK[tid]), 0, 0);
  }
  __builtin_amdgcn_s_wait_asynccnt(0);
  for (unsigned p = tid; p < nPix; p += THREADS) {
    const float r = sR[p];
    int idx = (int)ceilf(r) - 1;
    idx = idx < 0 ? 0 : (idx > (RADIUS - 1) ? (RADIUS - 1) : idx);
    const float tv = ws[idx]; /* table resident in L2 */
    const float h = ((r <= (float)RADIUS) ? tv : 0.0f) + sN[p];
    v2f e;
    e.x = h;
    e.y = sA[p];
    sPix[p] = e;
  }
#else
  for (unsigned p = tid; p < nPix; p += THREADS) {
    const unsigned gp = pixStart + p;
    const float r = rdist[gp];
    int idx = (int)ceilf(r) - 1;
    idx = idx < 0 ? 0 : (idx > (RADIUS - 1) ? (RADIUS - 1) : idx);
    const float tv = ws[idx];
    const float h = ((r <= (float)RADIUS) ? tv : 0.0f) + noise[gp];
    v2f e;
    e.x = h;
    e.y = aperture[gp];
    sPix[p] = e;
  }
  if (tid < N_WL) sK[tid] = ws[RADIUS + tid];
#endif
  __syncthreads();

  /* Stream the field: 4 elements (16B) per thread per iteration, NT hints. */
  const unsigned f4base = (unsigned)blockIdx.x * F4_PER_BLOCK + tid;
#pragma unroll
  for (int it = 0; it < ITERS; ++it) {
    const unsigned f4 = f4base + (unsigned)it * THREADS;
    const unsigned e0 = f4 * 4u;
    const v4f fr = __builtin_nontemporal_load((const v4f *)(fieldR + e0));
    const v4f fi = __builtin_nontemporal_load((const v4f *)(fieldI + e0));

    unsigned pix = e0 / 31u; /* magic-multiply division */
    unsigned wl = e0 - pix * 31u;
    unsigned lp = pix - pixStart;

    v4f orv, oiv;
#pragma unroll
    for (int j = 0; j < 4; ++j) {
      const v2f ha = sPix[lp];
      const float ph = sK[wl] * ha.x;
      const float s = __sinf(ph);
      const float c = __cosf(ph);
      orv[j] = (fr[j] * c - fi[j] * s) * ha.y;
      oiv[j] = (fr[j] * s + fi[j] * c) * ha.y;
      if (++wl == N_WL) {
        wl = 0;
        ++lp;
      }
    }
    __builtin_nontemporal_store(orv, (v4f *)(outR + e0));
    __builtin_nontemporal_store(oiv, (v4f *)(outI + e0));
  }
}

extern "C" void kernel_launch(void *const *d_in, const int *in_sizes, int n_in,
                              void *d_out, int out_size, void *d_ws,
                              size_t ws_size, hipStream_t stream) {
  (void)in_sizes;
  (void)n_in;
  (void)out_size;
  (void)ws_size;
  const float *hmw = (const float *)d_in[0];        /* height_map_weight[512] */
  const float *fieldR = (const float *)d_in[1];     /* field_real */
  const float *fieldI = (const float *)d_in[2];     /* field_imag */
  const float *wavelength = (const float *)d_in[3]; /* wavelength[31] */
  const float *noise = (const float *)d_in[4];      /* noise[1M] */
  const float *rdist = (const float *)d_in[5];      /* radius_distance[1M] */
  const float *aperture = (const float *)d_in[6];   /* aperture[1M] */

  float *ws = (float *)d_ws; /* [0,512): table, [512,543): k*delta_n */
  float *outR = (float *)d_out;
  float *outI = outR + (size_t)N_ELEM;

  doe_prep_kernel<<<1, 512, 0, stream>>>(hmw, wavelength, ws);
  doe_main_kernel<<<NBLOCKS, THREADS, 0, stream>>>(
      fieldR, fieldI, noise, rdist, aperture, ws, outR, outI);
}